// GaussianMomentDescriptor_77120432767609
// MI455X (gfx1250) — compile-verified
//
#include <hip/hip_runtime.h>
#include <hip/hip_bf16.h>
#include <math.h>

#define N_RADIAL   5
#define N_BASIS    7
#define R_MAX      6.0f
#define R_MIN      0.5f
#define N_SPECIES  119
#define M_PER_ATOM 200   // m0(5) + m1(15) + m2(45) + m3(135)
#define OUT_PER_ATOM 360 // 5 + 15 + 15 + 15 + 35 + 75 + 75 + 125
#define G_COLS     40    // 39 used + 1 zero pad (K = 10 * 4)

typedef __attribute__((ext_vector_type(2))) float v2f;
typedef __attribute__((ext_vector_type(8))) float v8f;

// lower-triangle index tables (match reference TRIL2 / TRIL3 ordering)
__constant__ int c_t2i[15] = {0,1,1,2,2,2,3,3,3,3,4,4,4,4,4};
__constant__ int c_t2j[15] = {0,0,1,0,1,2,0,1,2,3,0,1,2,3,4};
__constant__ int c_t3i[35] = {0, 1,1,1, 2,2,2,2,2,2, 3,3,3,3,3,3,3,3,3,3,
                              4,4,4,4,4,4,4,4,4,4,4,4,4,4,4};
__constant__ int c_t3j[35] = {0, 0,1,1, 0,1,1,2,2,2, 0,1,1,2,2,2,3,3,3,3,
                              0,1,1,2,2,2,3,3,3,3,4,4,4,4,4};
__constant__ int c_t3k[35] = {0, 0,0,1, 0,0,1,0,1,2, 0,0,1,0,1,2,0,1,2,3,
                              0,0,1,0,1,2,0,1,2,3,0,1,2,3,4};

// ---------------------------------------------------------------------------
// Kernel 1: zero the per-atom moment workspace (must be re-zeroed every call)
// ---------------------------------------------------------------------------
__global__ void gm_zero(float* __restrict__ m, int n) {
    int i = blockIdx.x * blockDim.x + threadIdx.x;
    if (i < n) m[i] = 0.0f;
}

// ---------------------------------------------------------------------------
// Kernel 2: neighbor phase — radial basis + direction moments, scattered with
// native f32 global atomics into the L2-resident moment buffer.
// ---------------------------------------------------------------------------
__global__ __launch_bounds__(256)
void gm_neighbor_scatter(const float* __restrict__ dr_vec,
                         const int*   __restrict__ Z,
                         const int*   __restrict__ nbr,
                         const float* __restrict__ W,
                         float*       __restrict__ m,
                         int n_neigh) {
    int n = blockIdx.x * blockDim.x + threadIdx.x;
    if (n >= n_neigh) return;

    const int ii = nbr[n];
    const int jj = nbr[n_neigh + n];
    if (ii == jj) return;                       // (idx_i != idx_j) mask

    const float dx = dr_vec[3*n+0];
    const float dy = dr_vec[3*n+1];
    const float dz = dr_vec[3*n+2];
    const float dr = sqrtf(dx*dx + dy*dy + dz*dz);
    if (dr >= R_MAX) return;                    // cutoff mask -> zero contribution

    const float inv = 1.0f / (dr + 1e-5f);
    const float nx = dx*inv, ny = dy*inv, nz = dz*inv;

    const float betta    = (float)(N_BASIS*N_BASIS) / (R_MAX*R_MAX);
    const float pi       = 3.14159265358979323846f;
    const float rad_norm = powf(2.0f*betta/pi, 0.75f);   // constant-folded
    const float cutoff   = 0.5f * (__cosf(pi * dr / R_MAX) + 1.0f);

    const float* coeff = W + ((size_t)Z[ii]*N_SPECIES + (size_t)Z[jj]) * (N_RADIAL*N_BASIS);
    __builtin_prefetch(coeff, 0, 0);            // global_prefetch_b8 on W row

    float basis[N_BASIS];
#pragma unroll
    for (int k = 0; k < N_BASIS; ++k) {
        float s = R_MIN + (float)k * ((R_MAX - R_MIN) / (float)(N_BASIS - 1));
        float t = dr - s;
        basis[k] = rad_norm * __expf(-betta * t * t);
    }

    float radial[N_RADIAL];
#pragma unroll
    for (int r = 0; r < N_RADIAL; ++r) {
        float acc = 0.0f;
#pragma unroll
        for (int k = 0; k < N_BASIS; ++k) acc += basis[k] * coeff[r*N_BASIS + k];
        radial[r] = acc * cutoff;
    }

    // direction tensor products: dn (3), dn⊗dn (9), dn⊗dn⊗dn (27)
    float d1[3] = {nx, ny, nz};
    float d2[9];
    float d3[27];
#pragma unroll
    for (int i = 0; i < 3; ++i)
#pragma unroll
        for (int j = 0; j < 3; ++j)
            d2[i*3+j] = d1[i] * d1[j];
#pragma unroll
    for (int i = 0; i < 3; ++i)
#pragma unroll
        for (int j = 0; j < 3; ++j)
#pragma unroll
            for (int k = 0; k < 3; ++k)
                d3[i*9+j*3+k] = d2[i*3+j] * d1[k];

    float* mb = m + (size_t)jj * M_PER_ATOM;
#pragma unroll
    for (int r = 0; r < N_RADIAL; ++r) {
        const float rad = radial[r];
        unsafeAtomicAdd(mb + r, rad);                                // m0
#pragma unroll
        for (int i = 0; i < 3; ++i)
            unsafeAtomicAdd(mb + 5 + r*3 + i, rad * d1[i]);          // m1
#pragma unroll
        for (int q = 0; q < 9; ++q)
            unsafeAtomicAdd(mb + 20 + r*9 + q, rad * d2[q]);         // m2
#pragma unroll
        for (int q = 0; q < 27; ++q)
            unsafeAtomicAdd(mb + 65 + r*27 + q, rad * d3[q]);        // m3
    }
}

// ---------------------------------------------------------------------------
// Kernel 3: per-atom contractions. One wave32 per atom.
// Gram part (contr_1/2/3) via V_WMMA_F32_16X16X4_F32:
//   G (16x40): rows 0-4 = m1 (cols 0-2), rows 5-9 = m2 (cols 3-11),
//              rows 10-14 = m3 (cols 12-38); disjoint column supports
//   => G·Gᵀ is block-diagonal = the three 5x5 Grams exactly (full f32).
// G is materialized zero-padded in LDS, so every WMMA K-step is fed by ONE
// unconditional 8B-aligned ds_load_b64 per lane (A and B register contents
// are identical for a symmetric Gram in the 16x16x4 f32 layout).
// ---------------------------------------------------------------------------
__global__ __launch_bounds__(256)
void gm_atom_contract(const float* __restrict__ m,
                      float*       __restrict__ out,
                      int n_atoms) {
    __shared__ float sm[8][M_PER_ATOM];      // per-wave raw moments
    __shared__ float sG[8][16 * G_COLS];     // per-wave padded G (16x40)
    __shared__ float sS[8][256];             // per-wave 16x16 Gram result

    const int lane = threadIdx.x & 31;
    const int w    = threadIdx.x >> 5;
    const int atom = blockIdx.x * 8 + w;
    const bool valid = (atom < n_atoms);

    if (valid) {
        // zero padded G (20 ds stores/lane)
#pragma unroll
        for (int t = lane; t < 16 * G_COLS; t += 32) sG[w][t] = 0.0f;
        // vectorized fetch of this atom's 200 moments (b128 loads)
        const float4* mg4 = (const float4*)(m + (size_t)atom * M_PER_ATOM);
        float4* sm4 = (float4*)sm[w];
        for (int t = lane; t < M_PER_ATOM / 4; t += 32) sm4[t] = mg4[t];
    }
    __syncthreads();

    if (valid) {
        // scatter moments 5..199 into their (row, k) slot of G
        for (int t = lane; t < 195; t += 32) {
            float v = sm[w][5 + t];
            int row, k;
            if (t < 15)      { row = t / 3;                k = t - row * 3; }
            else if (t < 60) { int q = t - 15; int d = q / 9;  row = 5 + d;  k = 3  + (q - d * 9);  }
            else             { int q = t - 60; int d = q / 27; row = 10 + d; k = 12 + (q - d * 27); }
            sG[w][row * G_COLS + k] = v;
        }
    }
    __syncthreads();

    if (valid) {
        const float* sGw = sG[w];
        const int row  = lane & 15;          // A: M index; B(=Gᵀ): N index
        const int koff = (lane >> 4) * 2;    // K sub-offset for upper half-wave
        v8f c = {};
#pragma unroll
        for (int kk = 0; kk < 10; ++kk) {    // K = 40
            // row*40 + kk*4 + koff is always even -> 8B-aligned ds_load_b64
            v2f a = *(const v2f*)(sGw + row * G_COLS + kk * 4 + koff);
            c = __builtin_amdgcn_wmma_f32_16x16x4_f32(
                    false, a, false, a, (short)0, c, false, false);
        }
        // C/D layout: lane L, vgpr v -> S[M = v + 8*(L>=16)][N = L&15]
        const int nn = lane & 15;
        const int mb = (lane >> 4) * 8;
#pragma unroll
        for (int v = 0; v < 8; ++v) sS[w][(mb + v) * 16 + nn] = c[v];
    }
    __syncthreads();
    if (!valid) return;

    const float* smw = sm[w];
    const float* Sg  = sS[w];
    const float* m1  = smw + 5;    // [5][3]
    const float* m2  = smw + 20;   // [5][3][3]
    const float* m3  = smw + 65;   // [5][3][3][3]
    float* ob = out + (size_t)atom * OUT_PER_ATOM;

    for (int o = lane; o < OUT_PER_ATOM; o += 32) {
        float val;
        if (o < 5) {                                   // contr_0 = m0
            val = smw[o];
        } else if (o < 20) {                           // contr_1 tril  (Gram block 0)
            int p = o - 5;
            val = Sg[c_t2i[p]*16 + c_t2j[p]];
        } else if (o < 35) {                           // contr_2 tril  (Gram block 1)
            int p = o - 20;
            val = Sg[(5 + c_t2i[p])*16 + 5 + c_t2j[p]];
        } else if (o < 50) {                           // contr_3 tril  (Gram block 2)
            int p = o - 35;
            val = Sg[(10 + c_t2i[p])*16 + 10 + c_t2j[p]];
        } else if (o < 85) {                           // contr_4 tril3
            int p = o - 50;
            int r = c_t3i[p], s = c_t3j[p], t = c_t3k[p];
            float acc = 0.0f;
            for (int j = 0; j < 3; ++j)
                for (int k = 0; k < 3; ++k) {
                    float b = 0.0f;
                    for (int i = 0; i < 3; ++i)
                        b += m2[r*9 + i*3 + j] * m2[s*9 + i*3 + k];
                    acc += b * m2[t*9 + j*3 + k];
                }
            val = acc;
        } else if (o < 160) {                          // contr_5 tril2 x 5
            int p = o - 85;
            int pr = p / 5, t = p % 5;
            int r = c_t2i[pr], s = c_t2j[pr];
            float acc = 0.0f;
            for (int i = 0; i < 3; ++i)
                for (int j = 0; j < 3; ++j)
                    acc += m1[r*3 + i] * m1[s*3 + j] * m2[t*9 + i*3 + j];
            val = acc;
        } else if (o < 235) {                          // contr_6 tril2 x 5
            int p = o - 160;
            int pr = p / 5, t = p % 5;
            int r = c_t2i[pr], s = c_t2j[pr];
            float acc = 0.0f;
            for (int k = 0; k < 3; ++k)
                for (int l = 0; l < 3; ++l) {
                    float cc = 0.0f;
                    for (int q = 0; q < 9; ++q)
                        cc += m3[r*27 + q*3 + k] * m3[s*27 + q*3 + l];
                    acc += cc * m2[t*9 + k*3 + l];
                }
            val = acc;
        } else {                                       // contr_7 full 5x5x5
            int p = o - 235;
            int r = p / 25, s = (p / 5) % 5, t = p % 5;
            float acc = 0.0f;
            for (int k = 0; k < 3; ++k) {
                float dsum = 0.0f;
                for (int q = 0; q < 9; ++q)
                    dsum += m3[r*27 + q*3 + k] * m2[s*9 + q];
                acc += dsum * m1[t*3 + k];
            }
            val = acc;
        }
        ob[o] = val;
    }
}

// ---------------------------------------------------------------------------
extern "C" void kernel_launch(void* const* d_in, const int* in_sizes, int n_in,
                              void* d_out, int out_size, void* d_ws, size_t ws_size,
                              hipStream_t stream) {
    const float* dr_vec = (const float*)d_in[0];
    const int*   Z      = (const int*)  d_in[1];
    const int*   nbr    = (const int*)  d_in[2];
    const float* W      = (const float*)d_in[3];
    float* out = (float*)d_out;
    float* mws = (float*)d_ws;   // n_atoms * 200 floats (16 MB)

    const int n_atoms = in_sizes[1];
    const int n_neigh = in_sizes[2] / 2;
    const int mcount  = n_atoms * M_PER_ATOM;

    gm_zero<<<(mcount + 255) / 256, 256, 0, stream>>>(mws, mcount);
    gm_neighbor_scatter<<<(n_neigh + 255) / 256, 256, 0, stream>>>(
        dr_vec, Z, nbr, W, mws, n_neigh);
    gm_atom_contract<<<(n_atoms + 7) / 8, 256, 0, stream>>>(mws, out, n_atoms);
}